// DenseEdgeConv_76811195121952
// MI455X (gfx1250) — compile-verified
//
#include <hip/hip_runtime.h>

// Problem constants (from reference): B=4, C=24, N=8192, G=24, K=16
#define Bn 4
#define Cn 24
#define Nn 8192
#define Gn 24
#define Kn 16

typedef __attribute__((ext_vector_type(2))) float v2f;
typedef __attribute__((ext_vector_type(4))) float v4f;
typedef __attribute__((ext_vector_type(8))) float v8f;

#define FINF 3.0e38f

// ---------------------------------------------------------------------------
// Kernel 1: transpose x (B,C,N) -> pts (B,N,C) row-major + squared norms.
// Coalesced reads per channel, contiguous 24-float rows for WMMA operand loads.
// ---------------------------------------------------------------------------
__global__ __launch_bounds__(256) void prep_kernel(const float* __restrict__ x,
                                                   float* __restrict__ pts,
                                                   float* __restrict__ rnorm) {
  int gid = blockIdx.x * blockDim.x + threadIdx.x;   // b*N + n
  if (gid >= Bn * Nn) return;
  int b = gid / Nn, n = gid - b * Nn;
  const float* xb = x + (size_t)b * Cn * Nn;
  float* prow = pts + (size_t)gid * Cn;
  float s = 0.f;
#pragma unroll
  for (int c = 0; c < Cn; ++c) {
    float v = xb[(size_t)c * Nn + n];
    prow[c] = v;
    s += v * v;
  }
  rnorm[gid] = s;
}

// ---------------------------------------------------------------------------
// Kernel 2: exact KNN via f32 WMMA Gram tiles + in-register incremental top-16.
// One wave per 16 query rows; 32 candidate columns (2 WMMA tiles) per loop iter.
//   dot:  12 x v_wmma_f32_16x16x4_f32 per iteration (C=24 -> 6 K-chunks each)
//   D:    d = rq + rc - 2*dot; self-distance poisoned to +INF (uniform branch)
//   scan: lane L owns row (L&15), 8 cols per tile (half L>>4); candidates read
//         back as 4x b128 LDS loads, min-reduced; insertion path only if the
//         tile can beat the current 16th-best (rare), merge of halves at end.
// ---------------------------------------------------------------------------
__global__ __launch_bounds__(32) void knn_kernel(const float* __restrict__ pts,
                                                 const float* __restrict__ rnorm,
                                                 int* __restrict__ idx_out) {
  __shared__ float dtile[16][32];   // cols 0-15: tile A, 16-31: tile B
  __shared__ float sdist[32][16];
  __shared__ int   sindx[32][16];

  const int lane  = threadIdx.x;
  const int b     = blockIdx.x >> 9;        // N/16 = 512 row-blocks per batch
  const int rb    = blockIdx.x & 511;
  const int col16 = lane & 15;
  const int kh    = lane >> 4;              // lane-half selects K offset (+2)
  const int qrow0 = rb * 16;

  const float* P  = pts   + (size_t)b * Nn * Cn;
  const float* RN = rnorm + (size_t)b * Nn;

  // A-matrix (queries), 16x4 f32 layout: lane holds A[m][kh*2 +0/1], m=lane&15
  v2f qa[6];
#pragma unroll
  for (int c = 0; c < 6; ++c)
    qa[c] = *(const v2f*)(P + (size_t)(qrow0 + col16) * Cn + c * 4 + kh * 2);

  // query norms for this lane's C/D rows (row m = i + kh*8)
  float rqv[8];
#pragma unroll
  for (int i = 0; i < 8; ++i) rqv[i] = RN[qrow0 + i + kh * 8];

  // incremental top-16 (unsorted, tracked max) kept in registers
  float kd[16];
  int   ki[16];
#pragma unroll
  for (int t = 0; t < 16; ++t) { kd[t] = FINF; ki[t] = -1; }
  float vmax = FINF;
  int   pmax = 0;

  const int cbase = kh * 8;          // column sub-range this lane scans

  for (int t2 = 0; t2 < Nn / 32; ++t2) {
    const int tA = t2 * 2, tB = t2 * 2 + 1;
    const int colA = tA * 16 + col16;
    const int colB = tB * 16 + col16;

    // two 16x16 Gram tiles, 6 K-chunks each
    v8f accA = {}, accB = {};
#pragma unroll
    for (int c = 0; c < 6; ++c) {
      v2f cbA = *(const v2f*)(P + (size_t)colA * Cn + c * 4 + kh * 2);
      v2f cbB = *(const v2f*)(P + (size_t)colB * Cn + c * 4 + kh * 2);
      accA = __builtin_amdgcn_wmma_f32_16x16x4_f32(false, qa[c], false, cbA,
                                                   (short)0, accA, false, false);
      accB = __builtin_amdgcn_wmma_f32_16x16x4_f32(false, qa[c], false, cbB,
                                                   (short)0, accB, false, false);
    }
    const float rcA = RN[colA];
    const float rcB = RN[colB];

    float dA[8], dB[8];
#pragma unroll
    for (int i = 0; i < 8; ++i) {
      dA[i] = rqv[i] + rcA - 2.0f * accA[i];
      dB[i] = rqv[i] + rcB - 2.0f * accB[i];
    }
    // poison self-distance (wave-uniform: happens for exactly one tile pair)
    if (tA == rb) {
#pragma unroll
      for (int i = 0; i < 8; ++i)
        if (qrow0 + i + kh * 8 == colA) dA[i] = FINF;
    }
    if (tB == rb) {
#pragma unroll
      for (int i = 0; i < 8; ++i)
        if (qrow0 + i + kh * 8 == colB) dB[i] = FINF;
    }
#pragma unroll
    for (int i = 0; i < 8; ++i) {
      dtile[i + kh * 8][col16]      = dA[i];
      dtile[i + kh * 8][16 + col16] = dB[i];
    }
    __syncthreads();

    // read back my 16 candidates as 4x float4, fast min-reject
    v4f q0 = *(const v4f*)&dtile[col16][cbase];
    v4f q1 = *(const v4f*)&dtile[col16][cbase + 4];
    v4f q2 = *(const v4f*)&dtile[col16][16 + cbase];
    v4f q3 = *(const v4f*)&dtile[col16][16 + cbase + 4];
    float vals[16];
#pragma unroll
    for (int j = 0; j < 4; ++j) {
      vals[j]      = q0[j];
      vals[4 + j]  = q1[j];
      vals[8 + j]  = q2[j];
      vals[12 + j] = q3[j];
    }
    float m16 = vals[0];
#pragma unroll
    for (int j = 1; j < 16; ++j) m16 = fminf(m16, vals[j]);

    if (m16 < vmax) {          // rare: something in this strip beats 16th-best
#pragma unroll
      for (int j = 0; j < 16; ++j) {
        const float d = vals[j];
        if (d < vmax) {
          const int cc = ((j < 8) ? tA : tB) * 16 + cbase + (j & 7);
#pragma unroll
          for (int t = 0; t < 16; ++t)
            if (t == pmax) { kd[t] = d; ki[t] = cc; }
          vmax = kd[0]; pmax = 0;
#pragma unroll
          for (int t = 1; t < 16; ++t)
            if (kd[t] > vmax) { vmax = kd[t]; pmax = t; }
        }
      }
    }
    __syncthreads();
  }

  // merge the two half-lists per row, emit ascending-sorted indices
#pragma unroll
  for (int t = 0; t < 16; ++t) { sdist[lane][t] = kd[t]; sindx[lane][t] = ki[t]; }
  __syncthreads();
  if (lane < 16) {
    const int r = lane;
    int* op = idx_out + ((size_t)b * Nn + qrow0 + r) * Kn;
    for (int s = 0; s < Kn; ++s) {
      float best = 3.4e38f; int bl = r, bt = 0;
#pragma unroll
      for (int t = 0; t < 16; ++t) {
        float v0 = sdist[r][t];      if (v0 < best) { best = v0; bl = r;      bt = t; }
        float v1 = sdist[r + 16][t]; if (v1 < best) { best = v1; bl = r + 16; bt = t; }
      }
      op[s] = sindx[bl][bt];
      sdist[bl][bt] = 3.4e38f;
    }
  }
}

// ---------------------------------------------------------------------------
// WMMA conv helper: D(24x16, padded to 2x16 row tiles) = W(24xIC) * F_in(ICx16)
// Bias folded into C operand; optional ReLU; result written back to F rows.
// ---------------------------------------------------------------------------
__device__ __forceinline__ void conv_wmma(const float* __restrict__ Ws, int IC,
                                          const float* __restrict__ biasv,
                                          float (*F)[16], int inbase, int outbase,
                                          bool relu, int lane) {
  const int col = lane & 15;
  const int kh  = lane >> 4;
  v8f acc0 = {}, acc1 = {};
#pragma unroll
  for (int i = 0; i < 8; ++i) {
    const int m0 = i + kh * 8;        // rows 0..15 (always valid, Gn=24)
    const int m1 = 16 + i + kh * 8;   // rows 16..31 (valid only < 24)
    acc0[i] = biasv[m0];
    acc1[i] = (m1 < Gn) ? biasv[m1] : 0.f;
  }
  const int nchunk = IC >> 2;
  for (int c = 0; c < nchunk; ++c) {
    const int ka = 4 * c + kh * 2;
    v2f a0 = *(const v2f*)&Ws[(size_t)col * IC + ka];
    v2f a1 = *(const v2f*)&Ws[(size_t)(col + 16) * IC + ka];
    v2f bb;
    bb.x = F[inbase + ka][col];
    bb.y = F[inbase + ka + 1][col];
    acc0 = __builtin_amdgcn_wmma_f32_16x16x4_f32(false, a0, false, bb,
                                                 (short)0, acc0, false, false);
    acc1 = __builtin_amdgcn_wmma_f32_16x16x4_f32(false, a1, false, bb,
                                                 (short)0, acc1, false, false);
  }
#pragma unroll
  for (int i = 0; i < 8; ++i) {
    const int m0 = i + kh * 8;
    float v0 = acc0[i]; if (relu) v0 = fmaxf(v0, 0.f);
    F[outbase + m0][col] = v0;
    const int m1 = 16 + i + kh * 8;
    if (m1 < Gn) {
      float v1 = acc1[i]; if (relu) v1 = fmaxf(v1, 0.f);
      F[outbase + m1][col] = v1;
    }
  }
}

// ---------------------------------------------------------------------------
// Kernel 3: gather neighbors, run 3 WMMA convs with concat-by-layout, max over k.
// Per-wave LDS feature matrix F[96][16]:
//   rows  0-23 : g1      rows 24-47 : g0
//   rows 48-71 : center  rows 72-95 : (knn - center), later overwritten by g2
// conv0 in = rows 48..95, conv1 in = rows 24..71, conv2 in = rows 0..71
// Output channels: [g2 | g1 | g0 | center], each max over the 16 neighbors.
// ---------------------------------------------------------------------------
__global__ __launch_bounds__(128) void edgeconv_kernel(
    const float* __restrict__ pts, const int* __restrict__ idx,
    const float* __restrict__ W0, const float* __restrict__ b0,
    const float* __restrict__ W1, const float* __restrict__ b1,
    const float* __restrict__ W2, const float* __restrict__ b2,
    float* __restrict__ out) {
  __shared__ float W0s[32 * 48];
  __shared__ float W1s[32 * 48];
  __shared__ float W2s[32 * 72];
  __shared__ float bias_s[3][24];
  __shared__ float Fs[4][96][16];

  const int tid = threadIdx.x;
  for (int t = tid; t < 32 * 48; t += 128) {
    int r = t / 48, c = t - r * 48;
    W0s[t] = (r < Gn) ? W0[r * 48 + c] : 0.f;
    W1s[t] = (r < Gn) ? W1[r * 48 + c] : 0.f;
  }
  for (int t = tid; t < 32 * 72; t += 128) {
    int r = t / 72, c = t - r * 72;
    W2s[t] = (r < Gn) ? W2[r * 72 + c] : 0.f;
  }
  if (tid < Gn) {
    bias_s[0][tid] = b0[tid];
    bias_s[1][tid] = b1[tid];
    bias_s[2][tid] = b2[tid];
  }
  __syncthreads();

  const int w    = tid >> 5;
  const int lane = tid & 31;
  const int pid  = blockIdx.x * 4 + w;       // 0 .. B*N-1
  const int b    = pid >> 13;                // N = 8192
  const int n    = pid & (Nn - 1);
  const float* P = pts + (size_t)b * Nn * Cn;
  float (*F)[16] = Fs[w];

  // build center + diff feature columns (one neighbor per lane 0..15)
  if (lane < 16) {
    const int ni = idx[((size_t)b * Nn + n) * Kn + lane];
    const float* cp = P + (size_t)n * Cn;
    const float* np = P + (size_t)ni * Cn;
#pragma unroll
    for (int c = 0; c < Cn; ++c) {
      float cv = cp[c], nv = np[c];
      F[48 + c][lane] = cv;        // center
      F[72 + c][lane] = nv - cv;   // knn - center
    }
  }
  __syncthreads();
  conv_wmma(W0s, 48, bias_s[0], F, /*in*/48, /*out*/24, /*relu*/true,  lane);
  __syncthreads();
  conv_wmma(W1s, 48, bias_s[1], F, /*in*/24, /*out*/ 0, /*relu*/true,  lane);
  __syncthreads();
  conv_wmma(W2s, 72, bias_s[2], F, /*in*/ 0, /*out*/72, /*relu*/false, lane);
  __syncthreads();

  // max over the 16 neighbors; 3 channels per lane (96 channels total)
#pragma unroll
  for (int q = 0; q < 3; ++q) {
    const int ch  = lane + q * 32;
    const int row = (ch < 24) ? (72 + ch) : (ch - 24);
    float m = F[row][0];
#pragma unroll
    for (int j = 1; j < Kn; ++j) m = fmaxf(m, F[row][j]);
    out[((size_t)b * 96 + ch) * Nn + n] = m;
  }
}

// ---------------------------------------------------------------------------
extern "C" void kernel_launch(void* const* d_in, const int* in_sizes, int n_in,
                              void* d_out, int out_size, void* d_ws, size_t ws_size,
                              hipStream_t stream) {
  const float* x  = (const float*)d_in[0];
  const float* W0 = (const float*)d_in[1];
  const float* b0 = (const float*)d_in[2];
  const float* W1 = (const float*)d_in[3];
  const float* b1 = (const float*)d_in[4];
  const float* W2 = (const float*)d_in[5];
  const float* b2 = (const float*)d_in[6];

  float* out   = (float*)d_out;
  float* pts   = (float*)d_ws;                       // B*N*C floats (3 MB)
  float* rnorm = pts + (size_t)Bn * Nn * Cn;         // B*N floats (128 KB)
  // idx output region follows y in d_out (i32 bits stored in the flat buffer)
  int* idx_out = (int*)(out + (size_t)Bn * 96 * Nn);

  prep_kernel<<<(Bn * Nn + 255) / 256, 256, 0, stream>>>(x, pts, rnorm);
  knn_kernel<<<Bn * (Nn / 16), 32, 0, stream>>>(pts, rnorm, idx_out);
  edgeconv_kernel<<<(Bn * Nn) / 4, 128, 0, stream>>>(pts, idx_out,
                                                     W0, b0, W1, b1, W2, b2, out);
}